// LogicConvSparseMatrix_6897717477609
// MI455X (gfx1250) — compile-verified
//
#include <hip/hip_runtime.h>
#include <cstdint>
#include <cstddef>

// Problem constants (from reference)
#define B_   16
#define C_   64
#define H_   128
#define W_   128
#define K_   128
#define OH_  126
#define OW_  126

#define TILE_R   16   // output rows per block tile
#define NTHREADS 256  // 8 waves (wave32)

// ---------------------------------------------------------------------------
// Fold the 15-term logic-weight sum into (alpha, beta, gamma, delta) per k:
//   out = alpha*a + beta*b + gamma*(a*b) + delta
// ---------------------------------------------------------------------------
__global__ void logic_coef_kernel(const float* __restrict__ w,
                                  float4* __restrict__ coef) {
    int k = threadIdx.x;
    if (k < K_) {
        const float* p = w + k * 16;
        float w1 = p[1],  w2 = p[2],  w3 = p[3],  w4 = p[4],  w5 = p[5];
        float w6 = p[6],  w7 = p[7],  w8 = p[8],  w9 = p[9],  w10 = p[10];
        float w11 = p[11], w12 = p[12], w13 = p[13], w14 = p[14], w15 = p[15];
        float alpha = w2 + w3 + w6 + w7 - w8 - w9 - w12 - w13;
        float beta  = w4 + w5 + w6 + w7 - w8 - w9 - w10 - w11;
        float gamma = w1 - w2 - w4 - 2.0f * w6 - w7 + w8 + 2.0f * w9
                      + w11 + w13 - w14;
        float delta = w8 + w9 + w10 + w11 + w12 + w13 + w14 + w15;
        coef[k] = make_float4(alpha, beta, gamma, delta);
    }
}

// ---------------------------------------------------------------------------
// Main kernel: one block handles one (b,k) plane x one 16-row output tile.
// Stage the two shifted input row-bands into LDS with CDNA5 async b128 copies
// (ASYNCcnt-tracked), then do the 4-op affine combine and NT-store the output.
// ---------------------------------------------------------------------------
__global__ __launch_bounds__(NTHREADS) void logic_conv_kernel(
    const float* __restrict__ x,
    const int*   __restrict__ pa,
    const int*   __restrict__ pb,
    const float4* __restrict__ coef,
    float* __restrict__ out) {

    __shared__ float lds_a[TILE_R * W_];   // 8 KB
    __shared__ float lds_b[TILE_R * W_];   // 8 KB

    const int p  = blockIdx.x;       // plane id = b*K + k
    const int b  = p >> 7;
    const int k  = p & (K_ - 1);
    const int r0 = blockIdx.y * TILE_R;
    const int t  = threadIdx.x;

    // Per-plane taps (wave-uniform -> scalar loads)
    const int ha = pa[3 * k + 0], wa = pa[3 * k + 1], ca = pa[3 * k + 2];
    const int hb = pb[3 * k + 0], wb = pb[3 * k + 1], cb = pb[3 * k + 2];
    const float4 cf = coef[k];

    const uint64_t xbase = (uint64_t)(uintptr_t)x;
    // Element offset of the first staged input row for each operand
    const int baseA = ((b * C_ + ca) * H_ + r0 + ha) * W_;
    const int baseB = ((b * C_ + cb) * H_ + r0 + hb) * W_;
    // Input rows actually available in this channel plane (OOB rows of the
    // tile are clamped to a safe in-bounds address and never consumed).
    const int rowsAvailA = H_ - (r0 + ha);
    const int rowsAvailB = H_ - (r0 + hb);

    // --- Async stage: 2 x (16 rows x 128 cols) f32 = 2 x 512 b128 chunks ---
#pragma unroll
    for (int j = 0; j < 2; ++j) {
        const int e4   = t + j * NTHREADS;  // 16-byte chunk id, 0..511
        const int elem = e4 * 4;            // float index within band
        {
            const int el = ((elem >> 7) < rowsAvailA) ? elem : 0;
            const uint32_t goff = (uint32_t)(baseA + el) * 4u;
            const uint32_t la = (uint32_t)(uintptr_t)(&lds_a[elem]);
            asm volatile("global_load_async_to_lds_b128 %0, %1, %2"
                         :: "v"(la), "v"(goff), "s"(xbase) : "memory");
        }
        {
            const int el = ((elem >> 7) < rowsAvailB) ? elem : 0;
            const uint32_t goff = (uint32_t)(baseB + el) * 4u;
            const uint32_t lb = (uint32_t)(uintptr_t)(&lds_b[elem]);
            asm volatile("global_load_async_to_lds_b128 %0, %1, %2"
                         :: "v"(lb), "v"(goff), "s"(xbase) : "memory");
        }
    }
    asm volatile("s_wait_asynccnt 0" ::: "memory");
    __syncthreads();

    // --- Compute: out = alpha*a + beta*b + gamma*a*b + delta ---
    const int rows = (OH_ - r0) < TILE_R ? (OH_ - r0) : TILE_R;
    const int npix = rows * OW_;
    float* outp = out + (size_t)p * (OH_ * OW_) + (size_t)r0 * OW_;

    for (int idx = t; idx < npix; idx += NTHREADS) {
        const int row = idx / OW_;
        const int col = idx - row * OW_;
        const float av = lds_a[row * W_ + col + wa];
        const float bv = lds_b[row * W_ + col + wb];
        float r = fmaf(av * bv, cf.z, cf.w);
        r = fmaf(av, cf.x, r);
        r = fmaf(bv, cf.y, r);
        // Output is write-once (130 MB): NT store protects L2 residency of x.
        __builtin_nontemporal_store(r, &outp[idx]);
    }
}

// ---------------------------------------------------------------------------
extern "C" void kernel_launch(void* const* d_in, const int* in_sizes, int n_in,
                              void* d_out, int out_size, void* d_ws,
                              size_t ws_size, hipStream_t stream) {
    const float* x   = (const float*)d_in[0];
    const int*   pa  = (const int*)d_in[1];
    const int*   pb  = (const int*)d_in[2];
    const float* wts = (const float*)d_in[3];
    float4* coef = (float4*)d_ws;   // 128 * 16 bytes = 2 KB scratch

    logic_coef_kernel<<<1, K_, 0, stream>>>(wts, coef);

    dim3 grid(B_ * K_, (OH_ + TILE_R - 1) / TILE_R);
    logic_conv_kernel<<<grid, NTHREADS, 0, stream>>>(x, pa, pb, coef,
                                                     (float*)d_out);
}